// QuantConv1dDepthwise_79182017069509
// MI455X (gfx1250) — compile-verified
//
#include <hip/hip_runtime.h>

// ---------------------------------------------------------------------------
// QuantConv1dDepthwise: depthwise causal conv1d (K=4) + bias + per-tensor
// int8 fake-quant.  B=8, C=2048, L=4096.
//
// Bandwidth-bound: ~0.6 GFLOP vs ~576-768 MB of HBM traffic (read x twice +
// write out once; pass 2 runs in reverse block order so its first ~192 MB of
// x reads hit the 192 MB global L2 left warm by pass 1; output stores are
// non-temporal so they don't evict x).  Target ~25-33 us at 23.3 TB/s.
//
// CDNA5 path: GLOBAL_LOAD_ASYNC_TO_LDS_B128 streams tiles (plus the 3-element
// causal halo) into LDS, synchronized with s_wait_asynccnt + barrier.
// ---------------------------------------------------------------------------

namespace {
constexpr int kB = 8;
constexpr int kC = 2048;
constexpr int kL = 4096;
constexpr int kTile = 2048;          // elements of one row per block
constexpr int kThreads = 256;        // 8 waves (wave32)
constexpr int kPerThread = kTile / kThreads;   // 8 outputs per thread
constexpr int kTilesPerRow = kL / kTile;       // 2
constexpr int kBlocks = kB * kC * kTilesPerRow; // 32768
}  // namespace

typedef int v4i __attribute__((ext_vector_type(4)));
typedef __attribute__((address_space(1))) v4i gv4i;   // global int4
typedef __attribute__((address_space(3))) v4i lv4i;   // LDS int4
typedef __attribute__((address_space(3))) void lvoid; // LDS void (for ptrtoint)
typedef float f4 __attribute__((ext_vector_type(4)));

__device__ __forceinline__ void async_copy_b128(const float* gsrc, float* ldst) {
#if __has_builtin(__builtin_amdgcn_global_load_async_to_lds_b128)
  __builtin_amdgcn_global_load_async_to_lds_b128((gv4i*)gsrc, (lv4i*)ldst, 0, 0);
#else
  unsigned lofs = (unsigned)(size_t)(lvoid*)ldst;          // 32-bit LDS address
  unsigned long long ga = (unsigned long long)(size_t)gsrc;
  asm volatile("global_load_async_to_lds_b128 %0, %1, off"
               :: "v"(lofs), "v"(ga) : "memory");
#endif
}

__device__ __forceinline__ void wait_async0() {
#if __has_builtin(__builtin_amdgcn_s_wait_asynccnt)
  __builtin_amdgcn_s_wait_asynccnt(0);
#else
  asm volatile("s_wait_asynccnt 0" ::: "memory");
#endif
}

// Stage one tile (+ 4-float halo slot) of a row into LDS via async DMA.
// lx[4 + i] == x[row_base + tile0 + i];  lx[1..3] are x[tile0-3..tile0-1]
// (zeros at row start, so the causal left-pad is implicit).
__device__ __forceinline__ void stage_tile(const float* __restrict__ x,
                                           long long row_base, int tile0,
                                           float* lx) {
  const int t = threadIdx.x;
  const float* g = x + row_base + tile0;
  async_copy_b128(g + 4 * t, lx + 4 + 4 * t);
  async_copy_b128(g + 4 * (t + kThreads), lx + 4 + 4 * (t + kThreads));
  if (t == 0) {
    if (tile0 > 0) {
      async_copy_b128(g - 4, lx);          // halo: previous 4 elements
    } else {
      lx[0] = 0.f; lx[1] = 0.f; lx[2] = 0.f; lx[3] = 0.f;  // causal zero pad
    }
  }
  wait_async0();     // our own async loads landed in LDS
  __syncthreads();   // everyone else's too
}

// Pass 1: conv+bias, reduce global abs-max into ws[0] (as float bits; all
// values are >= 0 so uint compare order == float order).
__global__ __launch_bounds__(kThreads) void conv_absmax_kernel(
    const float* __restrict__ x, const float* __restrict__ w,
    const float* __restrict__ bias, unsigned* __restrict__ absmax_bits) {
  __shared__ __align__(16) float lx[4 + kTile];
  __shared__ unsigned smax;

  const int bid = blockIdx.x;
  const int row = bid / kTilesPerRow;            // (b*C + c)
  const int tile0 = (bid % kTilesPerRow) * kTile;
  const int c = row % kC;
  const long long row_base = (long long)row * kL;
  const int t = threadIdx.x;

  if (t == 0) smax = 0u;
  stage_tile(x, row_base, tile0, lx);

  const float w0 = w[4 * c + 0], w1 = w[4 * c + 1];
  const float w2 = w[4 * c + 2], w3 = w[4 * c + 3];
  const float bz = bias[c];

  const int base = t * kPerThread;
  float m = 0.f;
#pragma unroll
  for (int i = 0; i < kPerThread; ++i) {
    // out[l] = sum_k w[k] * x[l-3+k]  ->  lx index (l_local + 1 + k)
    float acc = fmaf(w0, lx[base + i + 1],
                fmaf(w1, lx[base + i + 2],
                fmaf(w2, lx[base + i + 3],
                fmaf(w3, lx[base + i + 4], bz))));
    m = fmaxf(m, fabsf(acc));
  }
  atomicMax(&smax, __float_as_uint(m));  // ds_max_u32
  __syncthreads();
  if (t == 0) atomicMax(absmax_bits, smax);  // global_atomic_max_u32
}

// Pass 2: recompute conv+bias, fake-quant with the global scale, store.
// Blocks are mapped in REVERSE order so the first reads of pass 2 hit the
// tail of x, which is still resident in the 192 MB L2 after pass 1.
__global__ __launch_bounds__(kThreads) void conv_quant_kernel(
    const float* __restrict__ x, const float* __restrict__ w,
    const float* __restrict__ bias, const unsigned* __restrict__ absmax_bits,
    float* __restrict__ out) {
  __shared__ __align__(16) float lx[4 + kTile];

  const int bid = kBlocks - 1 - blockIdx.x;      // reverse traversal
  const int row = bid / kTilesPerRow;
  const int tile0 = (bid % kTilesPerRow) * kTile;
  const int c = row % kC;
  const long long row_base = (long long)row * kL;
  const int t = threadIdx.x;

  stage_tile(x, row_base, tile0, lx);

  const float amax = __uint_as_float(*absmax_bits);
  const float scale = fmaxf(amax, 1e-8f) * (1.0f / 127.0f);

  const float w0 = w[4 * c + 0], w1 = w[4 * c + 1];
  const float w2 = w[4 * c + 2], w3 = w[4 * c + 3];
  const float bz = bias[c];

  const int base = t * kPerThread;
  float r[kPerThread];
#pragma unroll
  for (int i = 0; i < kPerThread; ++i) {
    float acc = fmaf(w0, lx[base + i + 1],
                fmaf(w1, lx[base + i + 2],
                fmaf(w2, lx[base + i + 3],
                fmaf(w3, lx[base + i + 4], bz))));
    // jnp.round == round half to even == rintf (RNE default)
    float q = fminf(fmaxf(__builtin_rintf(acc / scale), -128.f), 127.f) * scale;
    r[i] = q;
  }
  float* o = out + row_base + tile0 + base;   // 16B aligned (base % 8 == 0)
  // Non-temporal: streaming output must not evict x from L2.
  __builtin_nontemporal_store(f4{r[0], r[1], r[2], r[3]}, (f4*)(o + 0));
  __builtin_nontemporal_store(f4{r[4], r[5], r[6], r[7]}, (f4*)(o + 4));
}

__global__ void init_ws_kernel(unsigned* p) { *p = 0u; }

extern "C" void kernel_launch(void* const* d_in, const int* in_sizes, int n_in,
                              void* d_out, int out_size, void* d_ws, size_t ws_size,
                              hipStream_t stream) {
  const float* x = (const float*)d_in[0];      // [B, C, L]
  const float* w = (const float*)d_in[1];      // [C, 1, 4]
  const float* b = (const float*)d_in[2];      // [C]
  float* out = (float*)d_out;                  // [B, C, L]
  unsigned* ws = (unsigned*)d_ws;              // 1 word: abs-max bits

  init_ws_kernel<<<1, 1, 0, stream>>>(ws);

  conv_absmax_kernel<<<kBlocks, kThreads, 0, stream>>>(x, w, b, ws);
  conv_quant_kernel<<<kBlocks, kThreads, 0, stream>>>(x, w, b, ws, out);
}